// diff_lpc2rc_19301583028962
// MI455X (gfx1250) — compile-verified
//
#include <hip/hip_runtime.h>

// ---------------------------------------------------------------------------
// diff_lpc2rc on MI455X (gfx1250)
//
// Per (b,t) vector of 16 floats: 15 steps of the backward Levinson recursion.
// Memory-bound streaming op (262 MB @ 23.3 TB/s ~= 11 us). Strategy:
//   * TDM (tensor_load_to_lds) stages each block's 16KB input tile into LDS,
//     using the TDM pad feature (1 dword pad per 16 dwords) so per-lane LDS
//     reads (stride 17 dwords) are bank-conflict free (gcd(17,64)==1).
//   * One lane per vector, recursion fully unrolled in VGPRs; reversal is a
//     static register permutation (free). WMMA is inapplicable: ki is
//     data-dependent per vector, so no shared matrix exists.
//   * v_rcp_f32 + 1 Newton step instead of the full IEEE divide sequence.
//   * Non-temporal b128 stores for the 131MB output stream.
// ---------------------------------------------------------------------------

typedef __attribute__((ext_vector_type(4))) unsigned int v4u;
typedef __attribute__((ext_vector_type(8))) int          v8i;
typedef __attribute__((ext_vector_type(4))) int          v4i;
typedef __attribute__((ext_vector_type(4))) float        f4;

#define NCOEF            16
#define VEC_PER_BLOCK    256
#define ELEMS_PER_BLOCK  (VEC_PER_BLOCK * NCOEF)   // 4096 floats = 16 KB
#define LDS_STRIDE       17                        // 16 data dwords + 1 pad dword

__global__ __launch_bounds__(VEC_PER_BLOCK) void
lpc2rc_tdm_kernel(const float* __restrict__ in, float* __restrict__ out, int n_vec)
{
    __shared__ float tile[VEC_PER_BLOCK * LDS_STRIDE];   // 17408 bytes

    const int       tid       = threadIdx.x;
    const long long base_elem = (long long)blockIdx.x * ELEMS_PER_BLOCK;
    const long long total     = (long long)n_vec * NCOEF;

    // ---- wave 0: issue one TDM DMA for the whole block's input tile --------
    if ((tid >> 5) == 0) {
        unsigned long long gaddr =
            (unsigned long long)(uintptr_t)in + (unsigned long long)base_elem * 4ull;
        unsigned int lds_base = (unsigned int)(uintptr_t)(void*)tile; // low 32 bits = LDS offset
        long long    rem      = total - base_elem;
        unsigned int dim0     = (rem >= (long long)ELEMS_PER_BLOCK)
                                  ? (unsigned int)ELEMS_PER_BLOCK
                                  : (unsigned int)rem;   // OOB tile rows read as zero

        // D# group 0: count=1 | lds_addr | global_addr[56:0] | type=2
        v4u g0;
        g0.x = 1u;
        g0.y = lds_base;
        g0.z = (unsigned int)gaddr;
        g0.w = (unsigned int)((gaddr >> 32) & 0x01FFFFFFull) | 0x80000000u;

        // D# group 1:
        //   word0: workgroup_mask=0, data_size=2 (4B), pad_enable=1,
        //          pad_interval=3 (16 dwords), pad_amount=0 (1 dword)
        //   tensor_dim0 = dim0, tensor_dim1 = 1, tile_dim0 = 4096, tile_dim1 = 1
        //   tensor_dim0_stride = 4096
        v8i g1;
        g1[0] = (int)((2u << 16) | (1u << 20) | (3u << 22));
        g1[1] = (int)((dim0 & 0xFFFFu) << 16);                 // abar=0 | dim0[15:0]
        g1[2] = (int)(((dim0 >> 16) & 0xFFFFu) | (1u << 16));  // dim0[31:16] | tdim1[15:0]=1
        g1[3] = (int)(((unsigned)ELEMS_PER_BLOCK) << 16);      // tdim1[31:16]=0 | tile_dim0
        g1[4] = 1;                                             // tile_dim1=1 | tile_dim2=0
        g1[5] = ELEMS_PER_BLOCK;                               // tensor_dim0_stride[31:0]
        g1[6] = 0;
        g1[7] = 0;

        v4i gz = {0, 0, 0, 0};
#if defined(__clang_major__) && (__clang_major__ >= 23)
        v8i gz8 = {0, 0, 0, 0, 0, 0, 0, 0};
        __builtin_amdgcn_tensor_load_to_lds(g0, g1, gz, gz, gz8, 0);
#else
        __builtin_amdgcn_tensor_load_to_lds(g0, g1, gz, gz, 0);
#endif
        __builtin_amdgcn_s_wait_tensorcnt(0);   // wave-private counter: issuer waits
    }
    __syncthreads();                            // publish LDS tile to all 8 waves

    // ---- per-lane recursion, fully in registers ----------------------------
    const float* __restrict__ src = &tile[tid * LDS_STRIDE];  // conflict-free banks
    float rc[NCOEF];
#pragma unroll
    for (int k = 0; k < NCOEF; ++k) rc[k] = src[k];

#pragma unroll
    for (int i = 1; i < NCOEF; ++i) {
        const int   m  = NCOEF - i;         // elements updated this step
        const float ki = rc[m];
        const float d  = 1.0f - ki * ki;
        float       r  = __builtin_amdgcn_rcpf(d);
        r = r * (2.0f - d * r);             // one Newton step: ~0.5 ulp
        float t[NCOEF];
#pragma unroll
        for (int k = 0; k < m; ++k) t[k] = (rc[k] - ki * rc[m - 1 - k]) * r;
#pragma unroll
        for (int k = 0; k < m; ++k) rc[k] = t[k];
    }

    // ---- non-temporal b128 stores (streaming output, don't thrash L2) ------
    const long long v = (long long)blockIdx.x * VEC_PER_BLOCK + tid;
    if (v < (long long)n_vec) {
        f4* dst = (f4*)(out + v * NCOEF);
        f4 o0 = {rc[0],  rc[1],  rc[2],  rc[3]};
        f4 o1 = {rc[4],  rc[5],  rc[6],  rc[7]};
        f4 o2 = {rc[8],  rc[9],  rc[10], rc[11]};
        f4 o3 = {rc[12], rc[13], rc[14], rc[15]};
        __builtin_nontemporal_store(o0, dst + 0);
        __builtin_nontemporal_store(o1, dst + 1);
        __builtin_nontemporal_store(o2, dst + 2);
        __builtin_nontemporal_store(o3, dst + 3);
    }
}

extern "C" void kernel_launch(void* const* d_in, const int* in_sizes, int n_in,
                              void* d_out, int out_size, void* d_ws, size_t ws_size,
                              hipStream_t stream)
{
    const float* in  = (const float*)d_in[0];   // (B, T, 16) fp32
    float*       out = (float*)d_out;           // (B, T, 16) fp32
    const int n_vec  = in_sizes[0] / NCOEF;     // B*T vectors
    const int blocks = (n_vec + VEC_PER_BLOCK - 1) / VEC_PER_BLOCK;
    lpc2rc_tdm_kernel<<<blocks, VEC_PER_BLOCK, 0, stream>>>(in, out, n_vec);
}